// WeightedGATConv_7086696038723
// MI455X (gfx1250) — compile-verified
//
#include <hip/hip_runtime.h>
#include <hip/hip_bf16.h>
#include <math.h>

typedef float v2f __attribute__((ext_vector_type(2)));
typedef float v8f __attribute__((ext_vector_type(8)));

#define HD 128          // NUM_HEADS * OUT_FEATS == IN_FEATS == 128
#define NHEAD 4
#define DOUT 32
#define NEG_SLOPE 0.2f

// ---------------------------------------------------------------------------
// 1) ft = feat @ fc_w.T  via V_WMMA_F32_16X16X4_F32 (full fp32 precision)
//    block = 256 threads = 8 waves; each wave computes one 16x16 output tile
//    (wave w -> column tile w); block covers 16 rows x 128 cols.
// ---------------------------------------------------------------------------
__global__ __launch_bounds__(256) void gat_gemm_wmma(
    const float* __restrict__ feat, const float* __restrict__ fcw,
    float* __restrict__ ft, int n_nodes) {
  const int lane = threadIdx.x & 31;
  const int wave = threadIdx.x >> 5;           // column tile 0..7
  const int row0 = blockIdx.x * 16;
  if (row0 >= n_nodes) return;
  const int half = lane >> 4;                  // K-pair selector
  const int r    = lane & 15;

  int arow = row0 + r;
  if (arow >= n_nodes) arow = n_nodes - 1;     // clamp loads for tail rows
  const int bcol = wave * 16 + r;              // output column (fc_w row)

  const float* aptr = feat + (size_t)arow * HD + 2 * half;
  const float* bptr = fcw  + (size_t)bcol * HD + 2 * half;

  v8f acc = {};
#pragma unroll
  for (int kc = 0; kc < HD / 4; ++kc) {
    v2f a = *(const v2f*)(aptr + kc * 4);      // A[M=arow][K=4kc+2h .. +1]
    v2f b = *(const v2f*)(bptr + kc * 4);      // B[K][N=bcol] == fc_w[bcol][K]
    acc = __builtin_amdgcn_wmma_f32_16x16x4_f32(
        /*neg_a=*/false, a, /*neg_b=*/false, b,
        /*c_mod=*/(short)0, acc, /*reuse_a=*/false, /*reuse_b=*/false);
  }

  // D layout: VGPR i -> M = 8*half + i, N = r (within tile)
#pragma unroll
  for (int i = 0; i < 8; ++i) {
    const int row = row0 + half * 8 + i;
    if (row < n_nodes) ft[(size_t)row * HD + wave * 16 + r] = acc[i];
  }
}

// ---------------------------------------------------------------------------
// 2) el/er: per (node, head) dot of ft row-slice with attn vectors
// ---------------------------------------------------------------------------
__global__ void gat_node_attn(const float* __restrict__ ft,
                              const float* __restrict__ attn_l,
                              const float* __restrict__ attn_r,
                              float* __restrict__ el, float* __restrict__ er,
                              int n_nodes) {
  const int t = blockIdx.x * blockDim.x + threadIdx.x;
  if (t >= n_nodes * NHEAD) return;
  const int h = t & (NHEAD - 1);
  const int n = t >> 2;
  const float* f  = ft + (size_t)n * HD + h * DOUT;
  const float* al = attn_l + h * DOUT;
  const float* ar = attn_r + h * DOUT;
  float sl = 0.f, sr = 0.f;
#pragma unroll
  for (int d = 0; d < DOUT; ++d) {
    const float v = f[d];
    sl += v * al[d];
    sr += v * ar[d];
  }
  el[t] = sl;
  er[t] = sr;
}

// ---------------------------------------------------------------------------
// 3) init: m = -inf, s = 0, out = bias (broadcast)  -- one pass over N*128
// ---------------------------------------------------------------------------
__global__ void gat_init(float* __restrict__ m, float* __restrict__ s,
                         float* __restrict__ out, const float* __restrict__ bias,
                         int n_nodes) {
  const int t = blockIdx.x * blockDim.x + threadIdx.x;
  const int nh = n_nodes * NHEAD;
  if (t < nh) { m[t] = -INFINITY; s[t] = 0.f; }
  if (t < n_nodes * HD) out[t] = bias[t & (HD - 1)];
}

// sign-split float atomic max (target starts at -inf)
__device__ __forceinline__ void atomicMaxF(float* addr, float val) {
  if (val >= 0.f)
    atomicMax((int*)addr, __float_as_int(val));
  else
    atomicMin((unsigned int*)addr, __float_as_uint(val));
}

// ---------------------------------------------------------------------------
// 4) edge logits + segment max into m
// ---------------------------------------------------------------------------
__global__ void gat_edge_logits(const int* __restrict__ src,
                                const int* __restrict__ dst,
                                const float* __restrict__ ew,
                                const float* __restrict__ el,
                                const float* __restrict__ er,
                                const float* __restrict__ wlin_w,
                                const float* __restrict__ wlin_b,
                                float* __restrict__ logits,
                                float* __restrict__ m, int n_edges) {
  const int t = blockIdx.x * blockDim.x + threadIdx.x;
  if (t >= n_edges * NHEAD) return;
  const int e = t >> 2;
  const int h = t & (NHEAD - 1);
  const int d = dst[e];
  float x = el[src[e] * NHEAD + h] + er[d * NHEAD + h];
  x = (x >= 0.f) ? x : NEG_SLOPE * x;          // leaky relu BEFORE weight
  const float w = ew[e] * wlin_w[0] + wlin_b[0];
  x *= w;
  logits[t] = x;
  atomicMaxF(&m[d * NHEAD + h], x);
}

// ---------------------------------------------------------------------------
// 5) exp(e - m[dst]) in place + segment sum into s
// ---------------------------------------------------------------------------
__global__ void gat_edge_exp(const int* __restrict__ dst,
                             float* __restrict__ logits,
                             const float* __restrict__ m,
                             float* __restrict__ s, int n_edges) {
  const int t = blockIdx.x * blockDim.x + threadIdx.x;
  if (t >= n_edges * NHEAD) return;
  const int e = t >> 2;
  const int h = t & (NHEAD - 1);
  const int d = dst[e];
  const float p = expf(logits[t] - m[d * NHEAD + h]);
  logits[t] = p;
  atomicAdd(&s[d * NHEAD + h], p);
}

// ---------------------------------------------------------------------------
// 6) scatter: one wave per edge; lane covers 4 channels (float4 gather,
//    4x global_atomic_add_f32 — output fits in 192MB L2, so atomics stay local)
// ---------------------------------------------------------------------------
__global__ __launch_bounds__(256) void gat_scatter(
    const int* __restrict__ src, const int* __restrict__ dst,
    const float* __restrict__ ft, const float* __restrict__ logits,
    const float* __restrict__ s, float* __restrict__ out, int n_edges) {
  const int lane = threadIdx.x & 31;
  const int wid  = (int)((blockIdx.x * blockDim.x + threadIdx.x) >> 5);
  const int nw   = (int)((gridDim.x * blockDim.x) >> 5);
  const int h    = lane >> 3;                   // head for channels 4*lane..+3

  for (int e = wid; e < n_edges; e += nw) {
    const int sn = src[e];
    const int dn = dst[e];
    // prefetch the next edge's source row (gfx1250 global_prefetch_b8)
    if (e + nw < n_edges)
      __builtin_prefetch(ft + (size_t)src[e + nw] * HD + lane * 4, 0, 0);

    const float a = logits[(size_t)e * NHEAD + h] / s[dn * NHEAD + h];
    const float4 f = *(const float4*)(ft + (size_t)sn * HD + lane * 4);
    float* o = out + (size_t)dn * HD + lane * 4;
    atomicAdd(o + 0, f.x * a);
    atomicAdd(o + 1, f.y * a);
    atomicAdd(o + 2, f.z * a);
    atomicAdd(o + 3, f.w * a);
  }
}

// ---------------------------------------------------------------------------
extern "C" void kernel_launch(void* const* d_in, const int* in_sizes, int n_in,
                              void* d_out, int out_size, void* d_ws, size_t ws_size,
                              hipStream_t stream) {
  const float* feat   = (const float*)d_in[0];
  const int*   src    = (const int*)  d_in[1];
  const int*   dst    = (const int*)  d_in[2];
  const float* ew     = (const float*)d_in[3];
  const float* fcw    = (const float*)d_in[4];
  const float* attn_l = (const float*)d_in[5];
  const float* attn_r = (const float*)d_in[6];
  const float* bias   = (const float*)d_in[7];
  const float* wlin_w = (const float*)d_in[8];
  const float* wlin_b = (const float*)d_in[9];
  float* out = (float*)d_out;

  const int n_nodes = in_sizes[0] / HD;
  const int n_edges = in_sizes[1];

  // workspace layout (floats)
  float* ws = (float*)d_ws;
  float* ft     = ws;                                   // n_nodes*128
  float* el     = ft + (size_t)n_nodes * HD;            // n_nodes*4
  float* er     = el + (size_t)n_nodes * NHEAD;
  float* mbuf   = er + (size_t)n_nodes * NHEAD;
  float* sbuf   = mbuf + (size_t)n_nodes * NHEAD;
  float* logits = sbuf + (size_t)n_nodes * NHEAD;       // n_edges*4

  // 1) WMMA projection
  {
    dim3 grid((n_nodes + 15) / 16);
    gat_gemm_wmma<<<grid, 256, 0, stream>>>(feat, fcw, ft, n_nodes);
  }
  // 2) attention halves
  {
    const int tot = n_nodes * NHEAD;
    gat_node_attn<<<(tot + 255) / 256, 256, 0, stream>>>(ft, attn_l, attn_r,
                                                         el, er, n_nodes);
  }
  // 3) init m/s/out
  {
    const int tot = n_nodes * HD;
    gat_init<<<(tot + 255) / 256, 256, 0, stream>>>(mbuf, sbuf, out, bias, n_nodes);
  }
  // 4) logits + segment max
  {
    const int tot = n_edges * NHEAD;
    gat_edge_logits<<<(tot + 255) / 256, 256, 0, stream>>>(
        src, dst, ew, el, er, wlin_w, wlin_b, logits, mbuf, n_edges);
  }
  // 5) exp + segment sum
  {
    const int tot = n_edges * NHEAD;
    gat_edge_exp<<<(tot + 255) / 256, 256, 0, stream>>>(dst, logits, mbuf,
                                                        sbuf, n_edges);
  }
  // 6) weighted scatter aggregation
  {
    gat_scatter<<<4096, 256, 0, stream>>>(src, dst, ft, logits, sbuf, out,
                                          n_edges);
  }
}